// RelativeLabelLoss_34119220199830
// MI455X (gfx1250) — compile-verified
//
#include <hip/hip_runtime.h>
#include <hip/hip_bf16.h>

#define GAMMA 0.2f
#define BLOCK 256
#define NWAVES (BLOCK / 32)
#define TILE_FLOATS 1024      // BLOCK lanes * 4 floats (one b128 per lane per tile)
#define DEPTH 8               // async pipeline depth (per-wave LDS buffers)

// ---------------------------------------------------------------------------
// CDNA5 async global->LDS copy (tracked with ASYNCcnt).
// GVS addressing: mem = SGPR64(base) + VGPR32(byte_off); vdst = LDS byte addr.
// Non-temporal: x is streamed exactly once -> don't let it churn L2.
// ---------------------------------------------------------------------------
__device__ __forceinline__ void async_load_f4(uint32_t lds_addr,
                                              const float* gbase,
                                              uint32_t byte_off) {
  asm volatile("global_load_async_to_lds_b128 %0, %1, %2 th:TH_LOAD_NT"
               :
               : "v"(lds_addr), "v"(byte_off), "s"(gbase)
               : "memory");
}

#define WAIT_ASYNC(n) asm volatile("s_wait_asynccnt " #n ::: "memory")

// Generic pointer into LDS -> 32-bit LDS byte address (ISA 10.2: the shared
// aperture carries the LDS offset in addr[31:0]).  Verified last round: the
// compiler folded this to the static LDS offset of the buffer.
__device__ __forceinline__ uint32_t lds_addr_u32(const void* p) {
  return (uint32_t)(uintptr_t)p;
}

// online-softmax merge of one value
__device__ __forceinline__ void osm1(float v, float& m, float& s) {
  float nm = fmaxf(m, v);
  s = s * __expf(m - nm) + __expf(v - nm);
  m = nm;
}

__global__ void __launch_bounds__(BLOCK)
relative_label_loss_rows(const float* __restrict__ x, const int* __restrict__ y,
                         int B, int C, int T, float* __restrict__ ws) {
  const int b    = blockIdx.x;
  const int tid  = threadIdx.x;
  const int wave = tid >> 5;
  const int lane = tid & 31;

  __shared__ __align__(16) float abuf[NWAVES * DEPTH * 128];  // 32 KB
  __shared__ float red_m[NWAVES], red_s[NWAVES];

  const float* row = x + (size_t)b * (size_t)C;

  const int numTiles = C / TILE_FLOATS;
  const int tailBeg  = numTiles * TILE_FLOATS;

  // per-lane LDS byte address of this wave's buffer 0 slot
  const uint32_t ldsBase =
      lds_addr_u32(&abuf[0]) + (uint32_t)((wave * DEPTH * 128 + lane * 4) * 4);

  float m = -__builtin_inff();
  float s = 0.0f;

  // ---- prologue: fill the async pipeline (DEPTH-1 tiles in flight) ----
  const int pre = numTiles < (DEPTH - 1) ? numTiles : (DEPTH - 1);
  for (int t = 0; t < pre; ++t)
    async_load_f4(ldsBase + (uint32_t)(t & (DEPTH - 1)) * 512u, row,
                  (uint32_t)(t * TILE_FLOATS + tid * 4) * 4u);

  // ---- steady state: issue tile t+7, wait for tile t, consume tile t ----
  // Async loads complete in order, so asynccnt <= (outstanding-1) implies the
  // oldest tile has fully landed in LDS.
  for (int t = 0; t < numTiles; ++t) {
    if (t + (DEPTH - 1) < numTiles) {
      const int tn = t + (DEPTH - 1);
      async_load_f4(ldsBase + (uint32_t)(tn & (DEPTH - 1)) * 512u, row,
                    (uint32_t)(tn * TILE_FLOATS + tid * 4) * 4u);
      WAIT_ASYNC(0x7);                       // 8 in flight -> oldest done
    } else {
      const int r = numTiles - 1 - t;        // remaining younger tiles
      switch (r) {
        case 6:  WAIT_ASYNC(0x6); break;
        case 5:  WAIT_ASYNC(0x5); break;
        case 4:  WAIT_ASYNC(0x4); break;
        case 3:  WAIT_ASYNC(0x3); break;
        case 2:  WAIT_ASYNC(0x2); break;
        case 1:  WAIT_ASYNC(0x1); break;
        default: WAIT_ASYNC(0x0); break;
      }
    }
    const float4 v = *(const float4*)&abuf[(wave * DEPTH + (t & (DEPTH - 1))) * 128 + lane * 4];
    // per-float4 local max -> 5 exps per 4 elements
    float nm = fmaxf(m, fmaxf(fmaxf(v.x, v.y), fmaxf(v.z, v.w)));
    s = s * __expf(m - nm) + __expf(v.x - nm) + __expf(v.y - nm) +
        __expf(v.z - nm) + __expf(v.w - nm);
    m = nm;
  }

  // ---- tail (C % 1024 elements) via plain loads ----
  for (int i = tailBeg + tid; i < C; i += BLOCK) osm1(row[i], m, s);

  // ---- wave32 shuffle reduction of (m, s) ----
  for (int off = 16; off > 0; off >>= 1) {
    float m2 = __shfl_xor(m, off, 32);
    float s2 = __shfl_xor(s, off, 32);
    float nm = fmaxf(m, m2);
    s = s * __expf(m - nm) + s2 * __expf(m2 - nm);
    m = nm;
  }
  if (lane == 0) { red_m[wave] = m; red_s[wave] = s; }
  __syncthreads();

  if (tid == 0) {
    float M = red_m[0], S = red_s[0];
    for (int w = 1; w < NWAVES; ++w) {
      float m2 = red_m[w], s2 = red_s[w];
      float nm = fmaxf(M, m2);
      S = S * __expf(M - nm) + s2 * __expf(m2 - nm);
      M = nm;
    }
    const float lse = M + __logf(S);
    const int* yrow = y + (size_t)b * T;

    // loss1 contribution: lse - x[b, y0]
    atomicAdd(&ws[0], lse - row[yrow[0]]);

    // ---- loss2 epilogue (<= 8 extra labels) ----
    int K = T - 1; if (K > 8) K = 8;
    int  safeArr[8];
    bool hasValid = false;
    float predMin = __builtin_inff();
    int   minJ = 0;
    for (int j = 0; j < K; ++j) {
      const int id   = yrow[1 + j];
      const bool val = (id != -1);
      const int safe = val ? id : 0;
      safeArr[j] = safe;
      const float pred = val ? row[safe] : __builtin_inff();
      if (pred < predMin) { predMin = pred; minJ = j; }   // first-min tie rule
      hasValid |= val;
    }
    if (hasValid) {
      const int rel = safeArr[minJ];
      // distinct excluded columns = {valid idx} \ {rel}
      int ex[8]; int ne = 0;
      for (int j = 0; j < K; ++j) {
        const int id = yrow[1 + j];
        if (id == -1 || id == rel) continue;
        bool dup = false;
        for (int k = 0; k < ne; ++k) if (ex[k] == id) { dup = true; break; }
        if (!dup) ex[ne++] = id;
      }
      float sExcl = 0.0f;
      for (int k = 0; k < ne; ++k) sExcl += __expf(row[ex[k]] - M);
      // c* = masked column whose rank in the masked subsequence equals rel
      int cstar = rel;
      for (int c = rel; c < C; ++c) {
        bool excl = false; int cnt = 0;
        for (int k = 0; k < ne; ++k) {
          if (ex[k] == c) excl = true;
          if (ex[k] <= c) ++cnt;
        }
        if (!excl && (c - cnt) == rel) { cstar = c; break; }
      }
      const float lseMasked = M + __logf(S - sExcl);
      atomicAdd(&ws[1], lseMasked - row[cstar]);
      atomicAdd(&ws[2], 1.0f);
    }
  }
}

__global__ void init_ws(float* ws) {
  if (threadIdx.x < 3) ws[threadIdx.x] = 0.0f;
}

__global__ void finalize(const float* __restrict__ ws, float* __restrict__ out, int B) {
  out[0] = ws[0] / (float)B + GAMMA * ws[1] / (1e-8f + ws[2]);
}

extern "C" void kernel_launch(void* const* d_in, const int* in_sizes, int n_in,
                              void* d_out, int out_size, void* d_ws, size_t ws_size,
                              hipStream_t stream) {
  const float* x = (const float*)d_in[0];
  const int*   y = (const int*)d_in[1];
  const int T = 5;                       // reference: y is (B, 5)
  const int B = in_sizes[1] / T;
  const int C = in_sizes[0] / B;
  float* ws = (float*)d_ws;

  hipLaunchKernelGGL(init_ws, dim3(1), dim3(32), 0, stream, ws);
  hipLaunchKernelGGL(relative_label_loss_rows, dim3(B), dim3(BLOCK), 0, stream,
                     x, y, B, C, T, ws);
  hipLaunchKernelGGL(finalize, dim3(1), dim3(1), 0, stream, ws, (float*)d_out, B);
}